// AttentionPooling_55697135894568
// MI455X (gfx1250) — compile-verified
//
#include <hip/hip_runtime.h>

// Problem constants (from reference): B=64, T=4096, D=256, U=64
#define TB 64
#define TT 4096
#define TD 256
#define TU 64
#define NEG_BIG (-1e9f)

typedef __attribute__((ext_vector_type(2))) float v2f;
typedef __attribute__((ext_vector_type(8))) float v8f;

// w staged in LDS as K-pairs: wlds[p][col] = (w[2p][col], w[2p+1][col]), p = 0..127
// pitch 80 float2 per row => lane-half offset 80*2 dwords == 32 mod 64 banks
#define WPAIRS 128
#define WP 80

// ---------------------------------------------------------------------------
// Kernel 0: zero the output (kernel 3 accumulates with atomics)
// ---------------------------------------------------------------------------
__global__ void ap_zero_out(float* __restrict__ out) {
  out[blockIdx.x * blockDim.x + threadIdx.x] = 0.0f;
}

// ---------------------------------------------------------------------------
// Kernel 1: masked attention scores via v_wmma_f32_16x16x4_f32 (native f32,
//   zero conversion cost; we are HBM-bound so f32 WMMA throughput is ample).
//   grid = (T/256, B), block = 256 threads (8 wave32s)
//   Each wave computes 2 row-tiles of 16 t-values; full U=64 via 4 N-tiles.
// ---------------------------------------------------------------------------
__global__ void __launch_bounds__(256)
ap_scores_wmma(const float* __restrict__ inp,   // [B,T,D]
               const int*   __restrict__ mask,  // [B,T]
               const float* __restrict__ w,     // [D,U]
               const float* __restrict__ bias,  // [U]
               const float* __restrict__ uvec,  // [U]
               float*       __restrict__ scores)// [B,T] (workspace)
{
  __shared__ float2 wlds[WPAIRS * WP];  // 80 KB

  const int b   = blockIdx.y;
  const int tid = threadIdx.x;

  // Stage w as K-pairs: wlds[p*WP + col] = (w[2p, col], w[2p+1, col])
  for (int idx = tid; idx < WPAIRS * TU; idx += 256) {
    int p   = idx >> 6;   // K-pair index (0..127)
    int col = idx & 63;   // u column
    wlds[p * WP + col] = make_float2(w[(2 * p) * TU + col],
                                     w[(2 * p + 1) * TU + col]);
  }
  __syncthreads();

  const int wave = tid >> 5;
  const int lane = tid & 31;
  const int hi   = lane >> 4;   // lane half (0: lanes 0-15, 1: lanes 16-31)
  const int n    = lane & 15;   // row (A) / col (B,C) index within tile

  // Per-lane bias and u-vector entries for the 4 N-tiles (column = ut*16 + n)
  float biasr[4], ur[4];
#pragma unroll
  for (int ut = 0; ut < 4; ++ut) {
    biasr[ut] = bias[ut * 16 + n];
    ur[ut]    = uvec[ut * 16 + n];
  }

  const int tblock = blockIdx.x * 256;

  for (int sub = 0; sub < 2; ++sub) {
    const int t0 = tblock + (wave * 2 + sub) * 16;

    union Acc { v8f v; float f[8]; } acc[4];
#pragma unroll
    for (int ut = 0; ut < 4; ++ut)
      acc[ut].v = (v8f){0.f, 0.f, 0.f, 0.f, 0.f, 0.f, 0.f, 0.f};

    // A fragment addressing (16x4 f32 tile, ISA layout):
    //   lane = M row (= n for both halves), VGPR j holds K = kb + hi*2 + j
    const float* rowp = inp + ((size_t)b * TT + (size_t)(t0 + n)) * TD;

#pragma unroll 4
    for (int kc = 0; kc < 64; ++kc) {
      const int kb = kc * 4;

      const float2 av = *(const float2*)(rowp + kb + hi * 2);
      const v2f a = (v2f){av.x, av.y};

      // B fragment (4x16 f32): lane = N col, VGPR j holds K = kb + hi*2 + j
      //   -> exactly one staged K-pair: p = kc*2 + hi
#pragma unroll
      for (int ut = 0; ut < 4; ++ut) {
        const float2 bv = wlds[(kc * 2 + hi) * WP + ut * 16 + n];
        const v2f bb = (v2f){bv.x, bv.y};
        acc[ut].v = __builtin_amdgcn_wmma_f32_16x16x4_f32(
            /*neg_a=*/false, a, /*neg_b=*/false, bb,
            /*c_mod=*/(short)0, acc[ut].v,
            /*reuse_a=*/false, /*reuse_b=*/false);
      }
    }

    // uit = tanh(proj + bias); score contribution = uit * u[col]
    // C layout: lane half hi, VGPR r -> element (M = hi*8 + r, N = n)
    float partial[8];
#pragma unroll
    for (int r = 0; r < 8; ++r) partial[r] = 0.f;
#pragma unroll
    for (int ut = 0; ut < 4; ++ut) {
#pragma unroll
      for (int r = 0; r < 8; ++r) {
        float v = tanhf(acc[ut].f[r] + biasr[ut]);
        partial[r] += v * ur[ut];
      }
    }

    // Reduce over the 16 lanes (columns) of each half; lane n==0 holds result.
#pragma unroll
    for (int r = 0; r < 8; ++r) {
      float p = partial[r];
      p += __shfl_xor(p, 1, 16);
      p += __shfl_xor(p, 2, 16);
      p += __shfl_xor(p, 4, 16);
      p += __shfl_xor(p, 8, 16);
      if (n == 0) {
        int t = t0 + hi * 8 + r;
        float m = (float)mask[b * TT + t];
        scores[(size_t)b * TT + t] = p + (1.0f - m) * NEG_BIG;
      }
    }
  }
}

// ---------------------------------------------------------------------------
// Kernel 2: in-place max-shifted softmax over T for each batch row
//   grid = B, block = 256
// ---------------------------------------------------------------------------
__global__ void __launch_bounds__(256)
ap_softmax(float* __restrict__ sc)
{
  __shared__ float red[256];
  const int b = blockIdx.x;
  const int tid = threadIdx.x;
  float* s = sc + (size_t)b * TT;

  float mx = -3.4e38f;
  for (int i = tid; i < TT; i += 256) mx = fmaxf(mx, s[i]);
  red[tid] = mx; __syncthreads();
  for (int off = 128; off > 0; off >>= 1) {
    if (tid < off) red[tid] = fmaxf(red[tid], red[tid + off]);
    __syncthreads();
  }
  mx = red[0];
  __syncthreads();

  float sum = 0.f;
  for (int i = tid; i < TT; i += 256) {
    float e = __expf(s[i] - mx);
    s[i] = e;
    sum += e;
  }
  red[tid] = sum; __syncthreads();
  for (int off = 128; off > 0; off >>= 1) {
    if (tid < off) red[tid] += red[tid + off];
    __syncthreads();
  }
  float inv = 1.0f / red[0];
  __syncthreads();

  for (int i = tid; i < TT; i += 256) s[i] *= inv;
}

// ---------------------------------------------------------------------------
// Kernel 3: context[b,d] += sum_{t in chunk} attn[b,t] * inputs[b,t,d]
//   grid = (T/256, B), block = 256 (one column per thread, coalesced rows)
// ---------------------------------------------------------------------------
__global__ void __launch_bounds__(256)
ap_context(const float* __restrict__ inp,
           const float* __restrict__ attn,
           float* __restrict__ out)
{
  __shared__ float aw[256];
  const int b  = blockIdx.y;
  const int t0 = blockIdx.x * 256;
  const int tid = threadIdx.x;

  aw[tid] = attn[(size_t)b * TT + t0 + tid];
  __syncthreads();

  const float* base = inp + ((size_t)b * TT + t0) * TD + tid;
  float acc = 0.f;
#pragma unroll 4
  for (int t = 0; t < 256; ++t) acc += aw[t] * base[(size_t)t * TD];

  atomicAdd(&out[b * TD + tid], acc);
}

// ---------------------------------------------------------------------------
extern "C" void kernel_launch(void* const* d_in, const int* in_sizes, int n_in,
                              void* d_out, int out_size, void* d_ws, size_t ws_size,
                              hipStream_t stream) {
  const float* inp  = (const float*)d_in[0];  // [B,T,D] f32
  const int*   mask = (const int*)  d_in[1];  // [B,T]   i32
  const float* w    = (const float*)d_in[2];  // [D,U]   f32
  const float* bias = (const float*)d_in[3];  // [U]     f32
  const float* uvec = (const float*)d_in[4];  // [U,1]   f32
  float* out    = (float*)d_out;              // [B,D]   f32
  float* scores = (float*)d_ws;               // [B,T]   f32 workspace (1 MB)

  ap_zero_out<<<dim3(TB * TD / 256), 256, 0, stream>>>(out);
  ap_scores_wmma<<<dim3(TT / 256, TB), 256, 0, stream>>>(inp, mask, w, bias, uvec, scores);
  ap_softmax<<<dim3(TB), 256, 0, stream>>>(scores);
  ap_context<<<dim3(TT / 256, TB), 256, 0, stream>>>(inp, scores, out);
}